// PDPostProcessing_15547781612038
// MI455X (gfx1250) — compile-verified
//
#include <hip/hip_runtime.h>
#include <hip/hip_bf16.h>

typedef float v2f __attribute__((ext_vector_type(2)));
typedef float v8f __attribute__((ext_vector_type(8)));

#define N_DET 8192
#define TOPK 100
#define IOU_THR 0.3f
#define INV_LEN (1.0f / 128.0f)

// ---------------------------------------------------------------------------
// Kernel 1: dets = y/128 + anchors @ P  via V_WMMA_F32_16X16X4_F32.
// One wave32 computes a 16-row tile. A = anchors tile (16x2, K padded to 4),
// B = constant P columns, C = y/128 accumulator tile. Two WMMAs per tile:
// columns 0..15 and columns 16..17 (N padded). All loads are unconditional
// with clamped in-bounds addresses; lane predicates become v_cndmask selects
// so each load group forms a single s_clause (no per-element exec branches).
// ---------------------------------------------------------------------------
__global__ __launch_bounds__(256) void dets_wmma_kernel(
    const float* __restrict__ y, const float* __restrict__ anchors,
    float* __restrict__ dets) {
  const int lane = threadIdx.x & 31;
  const int wave = threadIdx.x >> 5;
  const int m0   = (blockIdx.x * 8 + wave) * 16;   // 16 rows per wave
  const int half = lane >> 4;                      // 0: lanes 0-15, 1: lanes 16-31
  const int lcol = lane & 15;

  // A matrix 16x4 f32: lanes 0-15 hold (K=0,K=1) for M=lane; lanes 16-31 hold
  // (K=2,K=3) = zero padding. Row m0+lcol is valid for both halves, so load
  // unconditionally and select.
  const float ax = anchors[2 * (m0 + lcol) + 0];
  const float ay = anchors[2 * (m0 + lcol) + 1];
  v2f a;
  a.x = half ? 0.f : ax;
  a.y = half ? 0.f : ay;

  // B1: P columns 0..15 (lanes 0-15 K=0/1; lanes 16-31 K=2/3 pad) — pure ALU.
  v2f b1;
  b1.x = (half == 0 && lcol != 2 && (lcol & 1) == 0) ? 1.f : 0.f;  // anchor-x
  b1.y = (half == 0 && lcol != 3 && (lcol & 1) == 1) ? 1.f : 0.f;  // anchor-y

  // C1 accumulator: y/128, VGPR r <-> row m0 + r + half*8, col = lcol
  v8f c1;
#pragma unroll
  for (int r = 0; r < 8; ++r) {
    const int row = m0 + r + half * 8;
    c1[r] = y[row * 18 + lcol] * INV_LEN;
  }

  v8f d1 = __builtin_amdgcn_wmma_f32_16x16x4_f32(
      false, a, false, b1, (short)0, c1, false, false);

#pragma unroll
  for (int r = 0; r < 8; ++r) {
    const int row = m0 + r + half * 8;
    dets[row * 18 + lcol] = d1[r];
  }

  // Second WMMA: columns 16..17 (cols 18..31 padding, never stored).
  v2f b2;
  b2.x = (half == 0 && lcol == 0) ? 1.f : 0.f;  // col 16: even -> anchor-x
  b2.y = (half == 0 && lcol == 1) ? 1.f : 0.f;  // col 17: odd  -> anchor-y

  const bool valid = (lcol < 2);
  const int  ccol  = 16 + (valid ? lcol : 0);   // clamped: always in-bounds
  v8f c2;
#pragma unroll
  for (int r = 0; r < 8; ++r) {
    const int row = m0 + r + half * 8;
    const float v = y[row * 18 + ccol] * INV_LEN;  // unconditional load
    c2[r] = valid ? v : 0.f;                       // v_cndmask, no branch
  }
  v8f d2 = __builtin_amdgcn_wmma_f32_16x16x4_f32(
      false, a, false, b2, (short)0, c2, false, false);
  if (valid) {  // single masked store clause
#pragma unroll
    for (int r = 0; r < 8; ++r) {
      const int row = m0 + r + half * 8;
      dets[row * 18 + 16 + lcol] = d2[r];
    }
  }
}

// ---------------------------------------------------------------------------
// Kernel 2: scores = sigmoid(x); boxes (SoA) from dets cols 0..3;
// sort keys: descending score, ascending index tiebreak (stable argsort).
// ---------------------------------------------------------------------------
__global__ __launch_bounds__(256) void score_box_key_kernel(
    const float* __restrict__ x, const float* __restrict__ dets,
    float* __restrict__ scores, float* __restrict__ bx1,
    float* __restrict__ by1, float* __restrict__ bx2, float* __restrict__ by2,
    unsigned long long* __restrict__ keys) {
  const int i = blockIdx.x * 256 + threadIdx.x;
  if (i >= N_DET) return;
  const float s = 1.f / (1.f + expf(-x[i]));
  scores[i] = s;

  const float cx = dets[i * 18 + 0];
  const float cy = dets[i * 18 + 1];
  const float wh = dets[i * 18 + 2] * 0.5f;
  const float hh = dets[i * 18 + 3] * 0.5f;
  bx1[i] = cx - wh;
  by1[i] = cy - hh;
  bx2[i] = cx + wh;
  by2[i] = cy + hh;

  // monotone float->u32 (ascending), invert for descending-score sort
  unsigned u = __float_as_uint(s);
  u = (u & 0x80000000u) ? ~u : (u | 0x80000000u);
  keys[i] = ((unsigned long long)(~u) << 32) | (unsigned)i;
}

// ---------------------------------------------------------------------------
// Kernel 3: single-workgroup bitonic sort of 8192 u64 keys in dynamic LDS.
// ---------------------------------------------------------------------------
__global__ __launch_bounds__(1024) void sort_kernel(
    const unsigned long long* __restrict__ keys, unsigned* __restrict__ order) {
  extern __shared__ unsigned long long skeys[];
  const int N = N_DET;
  for (int i = threadIdx.x; i < N; i += 1024) skeys[i] = keys[i];
  __syncthreads();
  for (int k = 2; k <= N; k <<= 1) {
    for (int j = k >> 1; j > 0; j >>= 1) {
      for (int t = threadIdx.x; t < N / 2; t += 1024) {
        const int i = 2 * t - (t & (j - 1));
        const int p = i ^ j;  // partner (> i)
        const bool up = ((i & k) == 0);
        const unsigned long long a = skeys[i];
        const unsigned long long b = skeys[p];
        if (up ? (a > b) : (a < b)) {
          skeys[i] = b;
          skeys[p] = a;
        }
      }
      __syncthreads();
    }
  }
  for (int i = threadIdx.x; i < N; i += 1024)
    order[i] = (unsigned)(skeys[i] & 0xFFFFFFFFu);
}

// ---------------------------------------------------------------------------
// Kernel 4: LDS-resident greedy NMS (single workgroup) + output gather.
// Sorted boxes/areas live in LDS (~162 KB of the 320 KB WGP LDS); keep state
// is a 1 KB bitmask. Early-exit once TOPK boxes are kept (exact: greedy order
// means the first TOPK kept indices and the validity mask are already final).
// ---------------------------------------------------------------------------
__global__ __launch_bounds__(1024) void nms_kernel(
    const unsigned* __restrict__ order, const float* __restrict__ bx1,
    const float* __restrict__ by1, const float* __restrict__ bx2,
    const float* __restrict__ by2, const float* __restrict__ scores,
    const float* __restrict__ dets, float* __restrict__ out) {
  extern __shared__ float lds[];
  const int N = N_DET;
  float* sx1 = lds;
  float* sy1 = sx1 + N;
  float* sx2 = sy1 + N;
  float* sy2 = sx2 + N;
  float* sar = sy2 + N;
  unsigned* keepW = (unsigned*)(sar + N);      // N/32 words
  int* keptL = (int*)(keepW + N / 32);         // TOPK entries

  // Load boxes in score-sorted order into LDS (SoA), compute areas.
  for (int p = threadIdx.x; p < N; p += 1024) {
    const unsigned idx = order[p];
    const float a = bx1[idx], b = by1[idx], c = bx2[idx], d = by2[idx];
    sx1[p] = a; sy1[p] = b; sx2[p] = c; sy2[p] = d;
    sar[p] = (c - a) * (d - b);
  }
  for (int w = threadIdx.x; w < N / 32; w += 1024) keepW[w] = 0xFFFFFFFFu;
  __syncthreads();

  int count = 0;
  int i = 0;
  while (i < N) {
    // --- find next surviving candidate (uniform across all threads) ---
    int found = -1;
    while (i < N) {
      const unsigned wrd = keepW[i >> 5] & (0xFFFFFFFFu << (i & 31));
      if (wrd) { found = (i & ~31) + __ffs(wrd) - 1; break; }
      i = (i & ~31) + 32;
    }
    if (found < 0) break;
    i = found;
    if (threadIdx.x == 0) keptL[count] = i;
    count++;
    if (count >= TOPK) break;  // first TOPK kept + validity already final

    // --- suppress all j > i with IoU > threshold ---
    const float xi1 = sx1[i], yi1 = sy1[i], xi2 = sx2[i], yi2 = sy2[i];
    const float ai = sar[i];
#pragma unroll
    for (int q = 0; q < N / 1024; ++q) {
      const int j = q * 1024 + threadIdx.x;
      if (j > i) {
        float iw = fminf(xi2, sx2[j]) - fmaxf(xi1, sx1[j]);
        float ih = fminf(yi2, sy2[j]) - fmaxf(yi1, sy1[j]);
        iw = fmaxf(iw, 0.f);
        ih = fmaxf(ih, 0.f);
        const float inter = iw * ih;
        const float iou = inter / (ai + sar[j] - inter);
        if (iou > IOU_THR) atomicAnd(&keepW[j >> 5], ~(1u << (j & 31)));
      }
    }
    __syncthreads();
    i++;
  }
  __syncthreads();

  // --- output: 100 rows of [score, cx, cy, w, kp0x, kp0y, kp2x, kp2y] ---
  if (threadIdx.x < TOPK) {
    const int k = threadIdx.x;
    float o[8] = {0.f, 0.f, 0.f, 0.f, 0.f, 0.f, 0.f, 0.f};
    if (k < count) {
      const int p = keptL[k];
      const unsigned idx = order[p];
      o[0] = scores[idx];
      o[1] = dets[idx * 18 + 0];
      o[2] = dets[idx * 18 + 1];
      o[3] = dets[idx * 18 + 2];
      o[4] = dets[idx * 18 + 4];
      o[5] = dets[idx * 18 + 5];
      o[6] = dets[idx * 18 + 8];
      o[7] = dets[idx * 18 + 9];
    }
#pragma unroll
    for (int c = 0; c < 8; ++c) out[k * 8 + c] = o[c];
  }
}

// ---------------------------------------------------------------------------
extern "C" void kernel_launch(void* const* d_in, const int* in_sizes, int n_in,
                              void* d_out, int out_size, void* d_ws,
                              size_t ws_size, hipStream_t stream) {
  const float* x = (const float*)d_in[0];        // (1, 8192, 1)
  const float* y = (const float*)d_in[1];        // (1, 8192, 18)
  const float* anchors = (const float*)d_in[2];  // (8192, 2)
  float* out = (float*)d_out;                    // (100, 8)

  // Workspace carve-up (all offsets 8B-aligned).
  char* ws = (char*)d_ws;
  float* dets = (float*)(ws);                         // 8192*18 f32 = 589824 B
  float* bx1  = (float*)(ws + 589824);                // 32768 B each
  float* by1  = (float*)(ws + 589824 + 32768);
  float* bx2  = (float*)(ws + 589824 + 2 * 32768);
  float* by2  = (float*)(ws + 589824 + 3 * 32768);
  float* scores = (float*)(ws + 589824 + 4 * 32768);  // 720896
  unsigned long long* keys = (unsigned long long*)(ws + 753664 + 32768);
  unsigned* order = (unsigned*)(ws + 786432 + 65536); // 851968

  dets_wmma_kernel<<<N_DET / (16 * 8), 256, 0, stream>>>(y, anchors, dets);
  score_box_key_kernel<<<N_DET / 256, 256, 0, stream>>>(
      x, dets, scores, bx1, by1, bx2, by2, keys);
  sort_kernel<<<1, 1024, N_DET * sizeof(unsigned long long), stream>>>(keys,
                                                                       order);
  const size_t nms_lds =
      5 * N_DET * sizeof(float) + (N_DET / 32) * sizeof(unsigned) +
      TOPK * sizeof(int) + 64;
  nms_kernel<<<1, 1024, nms_lds, stream>>>(order, bx1, by1, bx2, by2, scores,
                                           dets, out);
}